// SwitchFeedForward_12575664243140
// MI455X (gfx1250) — compile-verified
//
#include <hip/hip_runtime.h>
#include <math.h>
#include <stdint.h>

// Problem constants from the reference
#define BBATCH 4
#define SSEQ 2048
#define TT (BBATCH*SSEQ)   // 8192 tokens
#define DD 512             // d_model
#define FF 2048            // d_ff
#define EE 8               // experts

// GEMM tiling: 256 threads = 8 wave32, waves in 2x4 grid, each wave owns a
// 32x32 region = 2x2 WMMA 16x16 tiles (A/B fragment reuse x2 each).
// Block tile 64x128, K staged in 32-deep LDS slabs, double buffered,
// filled by global_load_async_to_lds_b128 (ASYNCcnt pipeline).
#define BM 64
#define BN 128
#define KB 32
#define LDA (KB + 4)         // padded A row (floats); 144B row stride, 16B-aligned
#define LDB (BN + 4)         // padded B row (floats); 528B row stride, 16B-aligned
#define ABUFB (BM * LDA * 4) // bytes per A slab
#define BBUFB (KB * LDB * 4) // bytes per B slab

typedef float v2f __attribute__((ext_vector_type(2)));
typedef float v8f __attribute__((ext_vector_type(8)));

// ---- CDNA5 async global->LDS (bypasses VGPRs, tracked with ASYNCcnt) ----
__device__ __forceinline__ void async_ld_b128(uint32_t lds_addr, const float* gptr) {
  asm volatile("global_load_async_to_lds_b128 %0, %1, off"
               :: "v"(lds_addr), "v"(gptr)
               : "memory");
}
__device__ __forceinline__ void wait_async_le6() {  // prev slab done, next in flight
  asm volatile("s_wait_asynccnt 0x6" ::: "memory");
}
__device__ __forceinline__ void wait_async_0() {    // drain
  asm volatile("s_wait_asynccnt 0x0" ::: "memory");
}

// ---------------------------------------------------------------- utilities
__global__ void moe_zero_counts(int* __restrict__ cnt) {
  if (threadIdx.x < EE) cnt[threadIdx.x] = 0;
}

// ---------------------------------------------------------------- router
__global__ __launch_bounds__(256)
void moe_router(const float* __restrict__ x,
                const float* __restrict__ Ws,   // [D, E]
                const float* __restrict__ bs,   // [E]
                int*  __restrict__ cnt,         // [E]
                int*  __restrict__ perm,        // [E, T]
                float* __restrict__ scale) {    // [T]
  const int wave = threadIdx.x >> 5;
  const int lane = threadIdx.x & 31;
  const int t = blockIdx.x * 8 + wave;
  if (t >= TT) return;

  float acc[EE];
#pragma unroll
  for (int e = 0; e < EE; ++e) acc[e] = 0.0f;

  const float* xr = x + (size_t)t * DD;
  for (int k = lane; k < DD; k += 32) {
    const float xv = xr[k];
    const float* wr = Ws + (size_t)k * EE;
#pragma unroll
    for (int e = 0; e < EE; ++e) acc[e] = fmaf(xv, wr[e], acc[e]);
  }
#pragma unroll
  for (int e = 0; e < EE; ++e) {
#pragma unroll
    for (int off = 16; off > 0; off >>= 1)
      acc[e] += __shfl_down(acc[e], off, 32);
  }
  if (lane == 0) {
    float mx = -INFINITY; int arg = 0;
#pragma unroll
    for (int e = 0; e < EE; ++e) {
      const float l = acc[e] + bs[e];
      acc[e] = l;
      if (l > mx) { mx = l; arg = e; }
    }
    float ssum = 0.0f;
#pragma unroll
    for (int e = 0; e < EE; ++e) ssum += __expf(acc[e] - mx);
    scale[t] = 1.0f / ssum;              // top-1 prob = exp(0)/sum
    const int pos = atomicAdd(&cnt[arg], 1);
    perm[arg * TT + pos] = t;
  }
}

// ---------------------------------------------------------------- GEMM1
// h[tok,:] = gelu( x[tok,:] @ W1[e] + b1[e] ) for tok in expert e's list.
__global__ __launch_bounds__(256)
void moe_ffn1(const float* __restrict__ x,     // [T, D]
              const float* __restrict__ W1,    // [E, D, F]
              const float* __restrict__ b1,    // [E, F]
              const int*  __restrict__ cnt,
              const int*  __restrict__ perm,
              float* __restrict__ h) {         // [T, F]
  const int e  = blockIdx.z;
  const int m0 = blockIdx.x * BM;
  const int n0 = blockIdx.y * BN;
  const int count = cnt[e];
  if (m0 >= count) return;                     // block-uniform early exit

  __shared__ __align__(16) float ldsA[2][BM][LDA];
  __shared__ __align__(16) float ldsB[2][KB][LDB];
  __shared__ int ldsTok[BM];

  const int tid  = threadIdx.x;
  const int lane = tid & 31;
  const int wave = tid >> 5;
  const int wm   = (wave & 1) * 32;
  const int wn   = (wave >> 1) * 32;
  const int hl   = lane >> 4;
  const int l16  = lane & 15;

  if (tid < BM) {
    const int r = m0 + tid;
    ldsTok[tid] = (r < count) ? perm[e * TT + r] : -1;
  }
  __syncthreads();

  // A staging: 2 gathered rows x 1 float4 per thread per slab (token clamped,
  // not branched -> unconditional b128 async loads; dead rows never stored).
  const int rA0 = tid >> 3;                    // rows 0..31 (+32 for second)
  const int cA  = (tid & 7) * 4;               // 8 float4 per 32-float row
  const int tk0 = ldsTok[rA0], tk1 = ldsTok[rA0 + 32];
  const float* aG0 = x + (size_t)(tk0 < 0 ? 0 : tk0) * DD + cA;
  const float* aG1 = x + (size_t)(tk1 < 0 ? 0 : tk1) * DD + cA;
  const uint32_t aL0 = (uint32_t)(uintptr_t)&ldsA[0][rA0][cA];
  const uint32_t aL1 = (uint32_t)(uintptr_t)&ldsA[0][rA0 + 32][cA];

  // B staging: 4 float4 per thread per slab
  const float* Wb = W1 + (size_t)e * DD * FF;
  const float* bG[4];
  uint32_t bL[4];
#pragma unroll
  for (int j = 0; j < 4; ++j) {
    const int idx = tid + j * 256;
    const int r = idx >> 5;                    // 32 float4 per 128-float row
    const int c = (idx & 31) * 4;
    bG[j] = Wb + (size_t)r * FF + n0 + c;
    bL[j] = (uint32_t)(uintptr_t)&ldsB[0][r][c];
  }

  // issue slab 0 into buffer 0
  async_ld_b128(aL0, aG0);
  async_ld_b128(aL1, aG1);
#pragma unroll
  for (int j = 0; j < 4; ++j) async_ld_b128(bL[j], bG[j]);

  v8f cacc[2][2] = {{{0}, {0}}, {{0}, {0}}};
  const int nk = DD / KB;
  for (int i = 0; i < nk; ++i) {
    const int p = i & 1;
    if (i + 1 < nk) {                          // prefetch slab i+1 into buf 1-p
      const uint32_t qo = (uint32_t)(1 - p);
      const int k0 = (i + 1) * KB;
      async_ld_b128(aL0 + qo * ABUFB, aG0 + k0);
      async_ld_b128(aL1 + qo * ABUFB, aG1 + k0);
#pragma unroll
      for (int j = 0; j < 4; ++j)
        async_ld_b128(bL[j] + qo * BBUFB, bG[j] + (size_t)k0 * FF);
      wait_async_le6();                        // slab i landed; i+1 in flight
    } else {
      wait_async_0();
    }
    __syncthreads();

    const float (*Ap)[LDA] = ldsA[p];
    const float (*Bp)[LDB] = ldsB[p];
#pragma unroll
    for (int kk = 0; kk < KB; kk += 4) {
      v2f a[2], b[2];
      a[0].x = Ap[wm + l16][kk + 2*hl];        a[0].y = Ap[wm + l16][kk + 2*hl + 1];
      a[1].x = Ap[wm + 16 + l16][kk + 2*hl];   a[1].y = Ap[wm + 16 + l16][kk + 2*hl + 1];
      b[0].x = Bp[kk + 2*hl][wn + l16];        b[0].y = Bp[kk + 2*hl + 1][wn + l16];
      b[1].x = Bp[kk + 2*hl][wn + 16 + l16];   b[1].y = Bp[kk + 2*hl + 1][wn + 16 + l16];
#pragma unroll
      for (int ii = 0; ii < 2; ++ii)
#pragma unroll
        for (int jj = 0; jj < 2; ++jj)
          cacc[ii][jj] = __builtin_amdgcn_wmma_f32_16x16x4_f32(
              false, a[ii], false, b[jj], (short)0, cacc[ii][jj], false, false);
    }
    __syncthreads();                           // buf p free for slab i+2
  }

  // epilogue: bias + exact GELU, scatter by token id
#pragma unroll
  for (int ii = 0; ii < 2; ++ii) {
#pragma unroll
    for (int jj = 0; jj < 2; ++jj) {
      const int colg = n0 + wn + jj * 16 + l16;
      const float bias = b1[e * FF + colg];
#pragma unroll
      for (int v = 0; v < 8; ++v) {
        const int row = wm + ii * 16 + v + 8 * hl;
        const int tok = ldsTok[row];
        if (tok >= 0) {
          float val = cacc[ii][jj][v] + bias;
          val = 0.5f * val * (1.0f + erff(val * 0.70710678118654752f));
          h[(size_t)tok * FF + colg] = val;
        }
      }
    }
  }
}

// ---------------------------------------------------------------- GEMM2
// out[tok,:] = scale[tok] * ( h[tok,:] @ W2[e] + b2[e] )
__global__ __launch_bounds__(256)
void moe_ffn2(const float* __restrict__ h,     // [T, F]
              const float* __restrict__ W2,    // [E, F, D]
              const float* __restrict__ b2,    // [E, D]
              const int*  __restrict__ cnt,
              const int*  __restrict__ perm,
              const float* __restrict__ scale, // [T]
              float* __restrict__ out) {       // [T, D]
  const int e  = blockIdx.z;
  const int m0 = blockIdx.x * BM;
  const int n0 = blockIdx.y * BN;
  const int count = cnt[e];
  if (m0 >= count) return;

  __shared__ __align__(16) float ldsA[2][BM][LDA];
  __shared__ __align__(16) float ldsB[2][KB][LDB];
  __shared__ int   ldsTok[BM];
  __shared__ float ldsScl[BM];

  const int tid  = threadIdx.x;
  const int lane = tid & 31;
  const int wave = tid >> 5;
  const int wm   = (wave & 1) * 32;
  const int wn   = (wave >> 1) * 32;
  const int hl   = lane >> 4;
  const int l16  = lane & 15;

  if (tid < BM) {
    const int r = m0 + tid;
    const int tok = (r < count) ? perm[e * TT + r] : -1;
    ldsTok[tid] = tok;
    ldsScl[tid] = (tok >= 0) ? scale[tok] : 0.0f;
  }
  __syncthreads();

  // clamped token -> token 0's h row, which is always written (token 0 is
  // routed to exactly one expert), so async loads always hit valid data.
  const int rA0 = tid >> 3;
  const int cA  = (tid & 7) * 4;
  const int tk0 = ldsTok[rA0], tk1 = ldsTok[rA0 + 32];
  const float* aG0 = h + (size_t)(tk0 < 0 ? 0 : tk0) * FF + cA;
  const float* aG1 = h + (size_t)(tk1 < 0 ? 0 : tk1) * FF + cA;
  const uint32_t aL0 = (uint32_t)(uintptr_t)&ldsA[0][rA0][cA];
  const uint32_t aL1 = (uint32_t)(uintptr_t)&ldsA[0][rA0 + 32][cA];

  const float* Wb = W2 + (size_t)e * FF * DD;
  const float* bG[4];
  uint32_t bL[4];
#pragma unroll
  for (int j = 0; j < 4; ++j) {
    const int idx = tid + j * 256;
    const int r = idx >> 5;
    const int c = (idx & 31) * 4;
    bG[j] = Wb + (size_t)r * DD + n0 + c;
    bL[j] = (uint32_t)(uintptr_t)&ldsB[0][r][c];
  }

  async_ld_b128(aL0, aG0);
  async_ld_b128(aL1, aG1);
#pragma unroll
  for (int j = 0; j < 4; ++j) async_ld_b128(bL[j], bG[j]);

  v8f cacc[2][2] = {{{0}, {0}}, {{0}, {0}}};
  const int nk = FF / KB;
  for (int i = 0; i < nk; ++i) {
    const int p = i & 1;
    if (i + 1 < nk) {
      const uint32_t qo = (uint32_t)(1 - p);
      const int k0 = (i + 1) * KB;
      async_ld_b128(aL0 + qo * ABUFB, aG0 + k0);
      async_ld_b128(aL1 + qo * ABUFB, aG1 + k0);
#pragma unroll
      for (int j = 0; j < 4; ++j)
        async_ld_b128(bL[j] + qo * BBUFB, bG[j] + (size_t)k0 * DD);
      wait_async_le6();
    } else {
      wait_async_0();
    }
    __syncthreads();

    const float (*Ap)[LDA] = ldsA[p];
    const float (*Bp)[LDB] = ldsB[p];
#pragma unroll
    for (int kk = 0; kk < KB; kk += 4) {
      v2f a[2], b[2];
      a[0].x = Ap[wm + l16][kk + 2*hl];        a[0].y = Ap[wm + l16][kk + 2*hl + 1];
      a[1].x = Ap[wm + 16 + l16][kk + 2*hl];   a[1].y = Ap[wm + 16 + l16][kk + 2*hl + 1];
      b[0].x = Bp[kk + 2*hl][wn + l16];        b[0].y = Bp[kk + 2*hl + 1][wn + l16];
      b[1].x = Bp[kk + 2*hl][wn + 16 + l16];   b[1].y = Bp[kk + 2*hl + 1][wn + 16 + l16];
#pragma unroll
      for (int ii = 0; ii < 2; ++ii)
#pragma unroll
        for (int jj = 0; jj < 2; ++jj)
          cacc[ii][jj] = __builtin_amdgcn_wmma_f32_16x16x4_f32(
              false, a[ii], false, b[jj], (short)0, cacc[ii][jj], false, false);
    }
    __syncthreads();
  }

#pragma unroll
  for (int ii = 0; ii < 2; ++ii) {
#pragma unroll
    for (int jj = 0; jj < 2; ++jj) {
      const int colg = n0 + wn + jj * 16 + l16;
      const float bias = b2[e * DD + colg];
#pragma unroll
      for (int v = 0; v < 8; ++v) {
        const int row = wm + ii * 16 + v + 8 * hl;
        const int tok = ldsTok[row];
        if (tok >= 0) {
          out[(size_t)tok * DD + colg] = (cacc[ii][jj][v] + bias) * ldsScl[row];
        }
      }
    }
  }
}

// ---------------------------------------------------------------- launcher
extern "C" void kernel_launch(void* const* d_in, const int* in_sizes, int n_in,
                              void* d_out, int out_size, void* d_ws, size_t ws_size,
                              hipStream_t stream) {
  (void)in_sizes; (void)n_in; (void)out_size; (void)ws_size;
  const float* x  = (const float*)d_in[0];
  const float* Ws = (const float*)d_in[1];
  const float* bs = (const float*)d_in[2];
  const float* W1 = (const float*)d_in[3];
  const float* b1 = (const float*)d_in[4];
  const float* W2 = (const float*)d_in[5];
  const float* b2 = (const float*)d_in[6];
  float* out = (float*)d_out;

  // workspace layout (256B aligned segments): cnt[E] | perm[E*T] | scale[T] | h[T*F]
  char* ws = (char*)d_ws;
  int*   cnt   = (int*)ws;
  int*   perm  = (int*)(ws + 256);
  float* scale = (float*)(ws + 256 + (size_t)EE * TT * 4);
  float* h     = (float*)(ws + 256 + (size_t)EE * TT * 4 + (size_t)TT * 4);

  moe_zero_counts<<<1, 64, 0, stream>>>(cnt);
  moe_router<<<TT / 8, 256, 0, stream>>>(x, Ws, bs, cnt, perm, scale);

  dim3 g1(TT / BM, FF / BN, EE);
  moe_ffn1<<<g1, 256, 0, stream>>>(x, W1, b1, cnt, perm, h);

  dim3 g2(TT / BM, DD / BN, EE);
  moe_ffn2<<<g2, 256, 0, stream>>>(h, W2, b2, cnt, perm, scale, out);
}